// OT_Loss2_5171140624912
// MI455X (gfx1250) — compile-verified
//
#include <hip/hip_runtime.h>
#include <hip/hip_bf16.h>
#include <math.h>

typedef __attribute__((ext_vector_type(2))) float v2f;
typedef __attribute__((ext_vector_type(8))) float v8f;

#define M_BINS 4096     // 64x64 density bins
#define NPTS   512      // points per sample
#define NITER  100
#define REGC   10.0f

// ---------------------------------------------------------------------------
// Threefry-2x32 reproducing jax.random.randint(jax.random.key(1), (16,100), 0, 512)
// key(1) -> (k0,k1) = (0,1); counts = iota(1600) split into halves (c0=i, c1=i+800);
// out[0:800] = word0, out[800:1600] = word1; span=512 (pow2) -> idx = bits & 511.
// ---------------------------------------------------------------------------
__device__ __forceinline__ unsigned rotl32(unsigned x, unsigned r) {
    return (x << r) | (x >> (32u - r));
}

__device__ unsigned threefry_idx(int f) {
    unsigned c0, c1; int sel;
    if (f < 800) { c0 = (unsigned)f;        c1 = (unsigned)(f + 800); sel = 0; }
    else         { c0 = (unsigned)(f - 800); c1 = (unsigned)f;        sel = 1; }
    unsigned ks[3] = {0u, 1u, 0x1BD11BDBu};  // 0x1BD11BDA ^ 0 ^ 1
    unsigned x0 = c0 + ks[0];
    unsigned x1 = c1 + ks[1];
    const unsigned r0[4] = {13u, 15u, 26u, 6u};
    const unsigned r1[4] = {17u, 29u, 16u, 24u};
#pragma unroll
    for (int i = 0; i < 5; ++i) {
        const unsigned* rr = (i & 1) ? r1 : r0;
#pragma unroll
        for (int j = 0; j < 4; ++j) {
            x0 += x1; x1 = rotl32(x1, rr[j]); x1 ^= x0;
        }
        x0 += ks[(i + 1) % 3];
        x1 += ks[(i + 2) % 3] + (unsigned)(i + 1);
    }
    return (sel ? x1 : x0) & 511u;
}

// ---------------------------------------------------------------------------
// Block reduction: 256 threads = 8 wave32. is_max ? max : sum.
// ---------------------------------------------------------------------------
__device__ float block_reduce(float v, bool is_max, float* red, float* bc, int t) {
    __syncthreads();                 // protect red/bc across successive calls
#pragma unroll
    for (int o = 16; o > 0; o >>= 1) {
        float w = __shfl_xor(v, o, 32);
        v = is_max ? fmaxf(v, w) : (v + w);
    }
    if ((t & 31) == 0) red[t >> 5] = v;
    __syncthreads();
    if (t == 0) {
        float r = red[0];
        for (int i = 1; i < 8; ++i) r = is_max ? fmaxf(r, red[i]) : (r + red[i]);
        *bc = r;
    }
    __syncthreads();
    return *bc;
}

// ---------------------------------------------------------------------------
// Kernel A: per-sample sequential ASGD (100 iters, softmax over 4096 in regs),
// then beta-statistics (loss, ot_obj) and U = b*exp(beta/REG - bm) to workspace.
// Grid: 16 blocks (1/sample) x 256 threads. All state in registers.
// ---------------------------------------------------------------------------
__global__ __launch_bounds__(256) void ot_asgd_kernel(
    const float* __restrict__ normed, const float* __restrict__ unnormed,
    const float* __restrict__ pts, float* __restrict__ u_ws, float* __restrict__ out)
{
    __shared__ float red[8];
    __shared__ float bc;
    const int s = blockIdx.x;
    const int t = threadIdx.x;

    float b[16], lb[16], cur[16], ave[16];
#pragma unroll
    for (int e = 0; e < 16; ++e) {
        int j = e * 256 + t;
        float bv = normed[s * M_BINS + j];
        b[e] = bv; lb[e] = logf(bv);
        cur[e] = 0.0f; ave[e] = 0.0f;
    }
    const float a_w = 1.0f / 512.0f;            // a = 1/n
    const float lr  = 1.0f / (a_w / REGC);      // matches 1/max(a/REG)

    for (int k = 1; k <= NITER; ++k) {
        int i = (int)threefry_idx(s * NITER + (k - 1));
        float xi = pts[s * NPTS * 2 + i * 2 + 0];
        float yi = pts[s * NPTS * 2 + i * 2 + 1];

        float z[16];
        float zmax = -INFINITY;
#pragma unroll
        for (int e = 0; e < 16; ++e) {
            int j = e * 256 + t;
            float cy = (float)((j >> 6) * 8) + 4.0f;
            float cx = (float)((j & 63) * 8) + 4.0f;
            float dy = yi - cy, dx = xi - cx;
            float Mi = dy * dy + dx * dx;
            float zz = (cur[e] - Mi) / REGC + lb[e];
            z[e] = zz; zmax = fmaxf(zmax, zz);
        }
        float zm = block_reduce(zmax, true, red, &bc, t);

        float psum = 0.0f;
#pragma unroll
        for (int e = 0; e < 16; ++e) { z[e] = expf(z[e] - zm); psum += z[e]; }
        float Ssum = block_reduce(psum, false, red, &bc, t);

        float fk  = (float)k;
        float lrk = lr / sqrtf(fk);
        float c0  = 1.0f - 1.0f / fk;
#pragma unroll
        for (int e = 0; e < 16; ++e) {
            float khi = z[e] / Ssum;
            cur[e] = cur[e] + lrk * (b[e] - khi);
            ave[e] = cur[e] / fk + c0 * ave[e];
        }
    }

    // beta = ave.  Stats + U for kernel B.
    float sr[16];
    float sc_p = 0.0f, s1_p = 0.0f, ot_p = 0.0f, bm_p = -INFINITY;
#pragma unroll
    for (int e = 0; e < 16; ++e) {
        int j = e * 256 + t;
        float src = unnormed[s * M_BINS + j];
        sr[e] = src;
        sc_p += src;
        s1_p += src * ave[e];
        ot_p += b[e] * ave[e];
        bm_p  = fmaxf(bm_p, ave[e] / REGC + lb[e]);
    }
    float sc = block_reduce(sc_p, false, red, &bc, t);
    float S1 = block_reduce(s1_p, false, red, &bc, t);
    float ot = block_reduce(ot_p, false, red, &bc, t);
    float bm = block_reduce(bm_p, true,  red, &bc, t);

    float denom = sc * sc + 1e-16f;
    float g1 = sc / denom, g2 = S1 / denom;
    float l_p = 0.0f;
#pragma unroll
    for (int e = 0; e < 16; ++e) {
        int j = e * 256 + t;
        l_p += sr[e] * (g1 * ave[e] - g2);
        u_ws[s * M_BINS + j] = expf(ave[e] / REGC + lb[e] - bm);
    }
    float lsum = block_reduce(l_p, false, red, &bc, t);

    if (t == 0) {
        atomicAdd(out + 0, lsum);   // loss
        atomicAdd(out + 2, ot);     // ot_obj
    }
}

// ---------------------------------------------------------------------------
// Kernel B: wd via factorized transport plan with f32 WMMA.
//   P = GX * U^T, Q = HX * U^T   (512x64x64 per sample, V_WMMA_F32_16X16X4_F32)
//   s_i = <GY_i, P_i>, t_i = <HY_i, P_i> + <GY_i, Q_i>,  wd_i = a * t_i / s_i
// Grid: 16 blocks (1/sample) x 256 threads (8 waves); wave handles 16-row tiles.
// ---------------------------------------------------------------------------
__global__ __launch_bounds__(256) void ot_wd_wmma_kernel(
    const float* __restrict__ pts, const float* __restrict__ u_ws,
    float* __restrict__ out)
{
    __shared__ float Ut[M_BINS];   // transposed: Ut[jx*64 + jy] = U[jy][jx]
    const int s = blockIdx.x;
    const int t = threadIdx.x;

    for (int p = t; p < M_BINS; p += 256) {
        int jy = p >> 6, jx = p & 63;
        Ut[jx * 64 + jy] = u_ws[s * M_BINS + p];
    }
    __syncthreads();

    const int lane = t & 31;
    const int wv   = t >> 5;
    const int half = lane >> 4;   // A/B: K group; D: row half
    const int l16  = lane & 15;   // A: row, B/D: column

    float wd_acc = 0.0f;

    for (int rt = wv; rt < 32; rt += 8) {      // 4 row-tiles of 16 per wave
        const int r0 = rt * 16;

        // A-matrix source: x of row (r0 + l16)
        float xa = pts[s * NPTS * 2 + (r0 + l16) * 2 + 0];
        // D-layout rows for the tail dot: m = 8*half + v
        float yv[8];
#pragma unroll
        for (int v = 0; v < 8; ++v)
            yv[v] = pts[s * NPTS * 2 + (r0 + half * 8 + v) * 2 + 1];

        v8f accP[4] = {};
        v8f accQ[4] = {};

#pragma unroll
        for (int c = 0; c < 16; ++c) {         // K = 64 in steps of 4
            int kk = 4 * c + 2 * half;         // this lane's K pair (A and B share it)
            float cx0 = (float)(kk * 8 + 4), cx1 = cx0 + 8.0f;
            float dx0 = xa - cx0, dx1 = xa - cx1;
            float ga0 = expf(-dx0 * dx0 / REGC);
            float ga1 = expf(-dx1 * dx1 / REGC);
            v2f agx = {ga0, ga1};                         // GX slice (A)
            v2f ahx = {dx0 * dx0 * ga0, dx1 * dx1 * ga1}; // HX slice (A)
#pragma unroll
            for (int nt = 0; nt < 4; ++nt) {
                int n = nt * 16 + l16;
                v2f bu = {Ut[kk * 64 + n], Ut[(kk + 1) * 64 + n]}; // U^T slice (B)
                accP[nt] = __builtin_amdgcn_wmma_f32_16x16x4_f32(
                    false, agx, false, bu, (short)0, accP[nt], false, false);
                accQ[nt] = __builtin_amdgcn_wmma_f32_16x16x4_f32(
                    false, ahx, false, bu, (short)0, accQ[nt], false, false);
            }
        }

        // Tail: per-row dots with GY/HY using 16x16 f32 D layout
        // (lane = column n, VGPR v = row m%8, lanes>=16 hold rows 8..15).
        float accS[8] = {0}, accT[8] = {0};
#pragma unroll
        for (int nt = 0; nt < 4; ++nt) {
            int n = nt * 16 + l16;
            float cyn = (float)(n * 8 + 4);
#pragma unroll
            for (int v = 0; v < 8; ++v) {
                float dy = yv[v] - cyn;
                float gy = expf(-dy * dy / REGC);
                float hy = dy * dy * gy;
                accS[v] += gy * accP[nt][v];
                accT[v] += hy * accP[nt][v] + gy * accQ[nt][v];
            }
        }
#pragma unroll
        for (int v = 0; v < 8; ++v) {
#pragma unroll
            for (int o = 1; o < 16; o <<= 1) {
                accS[v] += __shfl_xor(accS[v], o, 32);
                accT[v] += __shfl_xor(accT[v], o, 32);
            }
        }
        if (l16 == 0) {
#pragma unroll
            for (int v = 0; v < 8; ++v) {
                float sden = fmaxf(accS[v], 1e-30f);
                wd_acc += (1.0f / 512.0f) * accT[v] / sden;
            }
        }
    }

    if (l16 == 0) atomicAdd(out + 1, wd_acc);  // wd (lanes 0 and 16 of each wave)
}

// ---------------------------------------------------------------------------
extern "C" void kernel_launch(void* const* d_in, const int* in_sizes, int n_in,
                              void* d_out, int out_size, void* d_ws, size_t ws_size,
                              hipStream_t stream) {
    (void)in_sizes; (void)n_in; (void)out_size; (void)ws_size;
    const float* normed   = (const float*)d_in[0];  // (16,1,64,64)
    const float* unnormed = (const float*)d_in[1];  // (16,1,64,64)
    const float* pts      = (const float*)d_in[2];  // (16,512,2)
    float* out  = (float*)d_out;                    // [loss, wd, ot_obj]
    float* u_ws = (float*)d_ws;                     // 16*4096 floats

    hipMemsetAsync(d_out, 0, 3 * sizeof(float), stream);
    ot_asgd_kernel<<<16, 256, 0, stream>>>(normed, unnormed, pts, u_ws, out);
    ot_wd_wmma_kernel<<<16, 256, 0, stream>>>(pts, u_ws, out);
}